// MeshCNN_29137058136334
// MI455X (gfx1250) — compile-verified
//
#include <hip/hip_runtime.h>
#include <hip/hip_bf16.h>
#include <stdint.h>

// Reference math degenerates: mu = h, var = 0  =>  z = (h-h)*rsqrt(eps) = 0
// => out[b,e,c] = beta[c] for all (b,e). The optimal kernel is a pure
// broadcast-store of the 16-float beta row over 1.6M rows (102.4 MB).
//
// CDNA5 path used: stage the 64B beta row in LDS, then stream it to global
// memory with GLOBAL_STORE_ASYNC_FROM_LDS_B128 (ASYNCcnt-tracked DMA, up to
// 63 in flight per wave), drained with s_wait_asynccnt.

#define THREADS 256

__global__ void __launch_bounds__(THREADS)
beta_broadcast_async(const float* __restrict__ beta,
                     float4* __restrict__ out,
                     long n4)
{
    // One 64-byte beta row staged in LDS (4 x float4 quarters).
    __shared__ float srow[16];
    if (threadIdx.x < 16) {
        srow[threadIdx.x] = beta[threadIdx.x];
    }
    __syncthreads();  // DS writes visible before async DMA reads LDS

    // Lane's global float4 index satisfies idx % 4 == threadIdx.x % 4 for
    // every iteration (block base and grid stride are multiples of 4), so the
    // LDS source address is loop-invariant per lane.
    // Generic LDS pointers carry the byte offset in their low 32 bits
    // (addr[31:0] per ISA aperture rules), so truncation yields the DS address.
    const uint32_t lds_addr =
        (uint32_t)(uintptr_t)(&srow[0]) + (uint32_t)(threadIdx.x & 3u) * 16u;

    long idx = (long)blockIdx.x * (long)blockDim.x + (long)threadIdx.x;
    const long stride = (long)gridDim.x * (long)blockDim.x;

    for (; idx < n4; idx += stride) {
        const float4* gp = out + idx;
        // vaddr = 64-bit global address (VGPR pair), vdata = LDS byte address.
        // Per-lane 16B transfer; one wave32 issue moves 512B, ASYNCcnt-tracked.
        asm volatile("global_store_async_from_lds_b128 %0, %1, off"
                     :
                     : "v"(gp), "v"(lds_addr)
                     : "memory");
    }

    // Drain outstanding async LDS->global transfers before wave exit.
    asm volatile("s_wait_asynccnt 0" ::: "memory");
}

extern "C" void kernel_launch(void* const* d_in, const int* in_sizes, int n_in,
                              void* d_out, int out_size, void* d_ws, size_t ws_size,
                              hipStream_t stream)
{
    // setup_inputs() order: x, W1, b1, W2, b2, W3, gamma, beta, elem_conn
    const float* beta = (const float*)d_in[7];
    float4* out = (float4*)d_out;

    // out_size = B*E*C = 25,600,000 floats -> 6,400,000 float4 stores.
    long n4 = (long)out_size / 4;

    // 1250 blocks x 256 threads = 320,000 lanes -> exactly 20 iterations per
    // lane (6,400,000 / 320,000), no tail; 8 wave32 per block fills WGPs.
    int grid = 1250;
    long total = (long)grid * THREADS;
    if (total > n4 && n4 > 0) {
        grid = (int)((n4 + THREADS - 1) / THREADS);  // safety for small outputs
    }

    beta_broadcast_async<<<grid, THREADS, 0, stream>>>(beta, out, n4);
}